// SRBF2d_44770739094122
// MI455X (gfx1250) — compile-verified
//
#include <hip/hip_runtime.h>

typedef float v2f __attribute__((ext_vector_type(2)));
typedef float v8f __attribute__((ext_vector_type(8)));

#define N1 1024
#define WAVES_PER_BLOCK 8
#define BLOCK (WAVES_PER_BLOCK * 32)
#define PTS_PER_BLOCK (WAVES_PER_BLOCK * 16)
#define L2E 1.4426950408889634f

// D-column assignment (accumulated over centers j with e = exp(-dx^2*d1 - dy^2*d2)):
//  set1 (ux,uy):  col0: d1*h      col1: d1*cx*h   col2: d2*h   col3: d2*cy*h
//  set2 (P,Px):   col4: h         col5: d1*h      col6: d1*cx*h
//  set3 (Q,Qy):   col7: h         col8: d2*h      col9: d2*cy*h
// then  ux = 2*(S1 - x*S0), uy = 2*(S3 - y*S2), P = S4, Px = 2*(S6 - x*S5),
//       Q = S7, Qy = 2*(S9 - y*S8).

__global__ __launch_bounds__(BLOCK) void srbf2d_wmma(
    const float* __restrict__ x,
    const float* __restrict__ h1, const float* __restrict__ c1, const float* __restrict__ w1,
    const float* __restrict__ h2, const float* __restrict__ c2, const float* __restrict__ w2,
    const float* __restrict__ h3, const float* __restrict__ c3, const float* __restrict__ w3,
    float* __restrict__ out, int Npts)
{
    __shared__ float4 sP[3 * N1];    // {cx, cy, -d1*log2e, -d2*log2e} per set (48 KB)
    __shared__ float  sB[10 * N1];   // B coefficients [j][col], stride 10 (40 KB);
                                     // reused as epilogue dump after barrier

    const int tid = threadIdx.x;

    // ---- stage center data + B-coefficient table into LDS ----
    for (int idx = tid; idx < 3 * N1; idx += BLOCK) {
        const int s = idx >> 10;
        const int j = idx & (N1 - 1);
        const float* c = (s == 0) ? c1 : (s == 1) ? c2 : c3;
        const float* w = (s == 0) ? w1 : (s == 1) ? w2 : w3;
        const float* h = (s == 0) ? h1 : (s == 1) ? h2 : h3;
        const float cx = c[2 * j], cy = c[2 * j + 1];
        const float wx = w[2 * j], wy = w[2 * j + 1];
        const float d1 = wx * wx, d2 = wy * wy;
        const float hv = h[j];
        sP[idx] = make_float4(cx, cy, -d1 * L2E, -d2 * L2E);
        float* brow = sB + j * 10;
        if (s == 0) {
            brow[0] = d1 * hv;
            brow[1] = d1 * cx * hv;
            brow[2] = d2 * hv;
            brow[3] = d2 * cy * hv;
        } else if (s == 1) {
            brow[4] = hv;
            brow[5] = d1 * hv;
            brow[6] = d1 * cx * hv;
        } else {
            brow[7] = hv;
            brow[8] = d2 * hv;
            brow[9] = d2 * cy * hv;
        }
    }
    __syncthreads();

    const int wave = tid >> 5;
    const int lane = tid & 31;
    const int n    = lane & 15;   // A row / B,D column index
    const int half = lane >> 4;   // which K pair this lane supplies

    const int tile  = blockIdx.x * PTS_PER_BLOCK + wave * 16;
    const int point = tile + n;
    const float px = x[2 * point];
    const float py = x[2 * point + 1];

    const int bIdx = (n < 10) ? n : 0;            // lanes n>=10 broadcast-read col 0
    const float set0m = (n < 4) ? 1.0f : 0.0f;    // ...and are zeroed by these masks
    const float set1m = (n >= 4 && n < 7) ? 1.0f : 0.0f;
    const float set2m = (n >= 7 && n < 10) ? 1.0f : 0.0f;

    v8f acc0 = {}, acc1 = {}, acc2 = {};

    #pragma unroll 2
    for (int j0 = 0; j0 < N1; j0 += 4) {
        const int jb = j0 + 2 * half;             // A layout: V0 -> K = 2*half, V1 -> K+1
        v2f a0 = {}, a1 = {}, a2 = {};
        v2f b0 = {}, b1 = {}, b2 = {};
        #pragma unroll
        for (int t = 0; t < 2; ++t) {
            const int j = jb + t;
            const float4 p0 = sP[j];
            const float4 p1 = sP[N1 + j];
            const float4 p2 = sP[2 * N1 + j];
            const float bt = sB[j * 10 + bIdx];

            const float dx0 = px - p0.x, dy0 = py - p0.y;
            const float e0 = __builtin_amdgcn_exp2f(fmaf(dx0 * dx0, p0.z, dy0 * dy0 * p0.w));
            const float dx1 = px - p1.x, dy1 = py - p1.y;
            const float e1 = __builtin_amdgcn_exp2f(fmaf(dx1 * dx1, p1.z, dy1 * dy1 * p1.w));
            const float dx2 = px - p2.x, dy2 = py - p2.y;
            const float e2 = __builtin_amdgcn_exp2f(fmaf(dx2 * dx2, p2.z, dy2 * dy2 * p2.w));

            a0[t] = e0; a1[t] = e1; a2[t] = e2;
            b0[t] = bt * set0m;
            b1[t] = bt * set1m;
            b2[t] = bt * set2m;
        }
        // D = A x B + C, full f32, 16x16x4; disjoint columns per accumulator
        acc0 = __builtin_amdgcn_wmma_f32_16x16x4_f32(false, a0, false, b0, (short)0, acc0, false, false);
        acc1 = __builtin_amdgcn_wmma_f32_16x16x4_f32(false, a1, false, b1, (short)0, acc1, false, false);
        acc2 = __builtin_amdgcn_wmma_f32_16x16x4_f32(false, a2, false, b2, (short)0, acc2, false, false);
    }

    v8f acc = acc0 + acc1 + acc2;   // columns are disjoint, sum merges them

    __syncthreads();                // all waves done reading sB as coefficients
    float* dump = sB + wave * 256;  // per-wave 16x16 transpose buffer (8 KB total)
    #pragma unroll
    for (int r = 0; r < 8; ++r) {
        const int m = r + 8 * half; // C/D layout: lanes 0-15 hold M=r, lanes 16-31 M=r+8
        dump[m * 16 + n] = acc[r];
    }
    __syncthreads();

    // epilogue: lane handles point row n (same point whose px/py it already holds)
    const float* row = dump + n * 16;
    if (half == 0) {
        const float S0 = row[0], S1 = row[1], S2 = row[2], S3 = row[3], S4 = row[4];
        out[0 * Npts + point] = 2.0f * (S1 - px * S0);   // ux
        out[1 * Npts + point] = 2.0f * (S3 - py * S2);   // uy
        out[2 * Npts + point] = S4;                      // P
    } else {
        const float S5 = row[5], S6 = row[6], S7 = row[7], S8 = row[8], S9 = row[9];
        out[3 * Npts + point] = 2.0f * (S6 - px * S5);   // Px
        out[4 * Npts + point] = S7;                      // Q
        out[5 * Npts + point] = 2.0f * (S9 - py * S8);   // Qy
    }
}

extern "C" void kernel_launch(void* const* d_in, const int* in_sizes, int n_in,
                              void* d_out, int out_size, void* d_ws, size_t ws_size,
                              hipStream_t stream) {
    const float* x  = (const float*)d_in[0];
    const float* h1 = (const float*)d_in[1];
    const float* c1 = (const float*)d_in[2];
    const float* w1 = (const float*)d_in[3];
    const float* h2 = (const float*)d_in[4];
    const float* c2 = (const float*)d_in[5];
    const float* w2 = (const float*)d_in[6];
    const float* h3 = (const float*)d_in[7];
    const float* c3 = (const float*)d_in[8];
    const float* w3 = (const float*)d_in[9];

    const int Npts = in_sizes[0] / 2;          // 65536
    dim3 grid(Npts / PTS_PER_BLOCK);           // 512 workgroups of 8 waves
    srbf2d_wmma<<<grid, BLOCK, 0, stream>>>(x, h1, c1, w1, h2, c2, w2,
                                            h3, c3, w3, (float*)d_out, Npts);
}